// EdgeConvBlock_22625887715370
// MI455X (gfx1250) — compile-verified
//
#include <hip/hip_runtime.h>
#include <hip/hip_bf16.h>
#include <math.h>

typedef float v2f __attribute__((ext_vector_type(2)));
typedef float v8f __attribute__((ext_vector_type(8)));
typedef unsigned int u32x4 __attribute__((ext_vector_type(4)));
typedef int i32x8 __attribute__((ext_vector_type(8)));
typedef int i32x4 __attribute__((ext_vector_type(4)));

constexpr int B   = 8;
constexpr int C   = 64;
constexpr int N   = 4096;
constexpr int O   = 64;
constexpr int KNN = 20;
constexpr int NPART = 256;

// workspace layout (float units)
constexpr size_t SQ_OFF   = 0;                               // B*N
constexpr size_t IDX_OFF  = SQ_OFF  + (size_t)B*N;           // B*N*KNN ints
constexpr size_t P_OFF    = IDX_OFF + (size_t)B*N*KNN;       // B*N*O
constexpr size_t Q_OFF    = P_OFF   + (size_t)B*N*O;         // B*N*O
constexpr size_t PART_OFF = Q_OFF   + (size_t)B*N*O;         // NPART*O*2
constexpr size_t ST_OFF   = PART_OFF + (size_t)NPART*O*2;    // 2*O

// ---------------------------------------------------------------------------
// Kernel 1: per-point squared norms  sq[b,n] = sum_c x[b,c,n]^2
// ---------------------------------------------------------------------------
__global__ void sq_kernel(const float* __restrict__ x, float* __restrict__ sq) {
  int g = blockIdx.x * 256 + threadIdx.x;      // over B*N
  int b = g / N, n = g % N;
  const float* xb = x + (size_t)b * C * N + n;
  float s = 0.f;
#pragma unroll
  for (int c = 0; c < C; ++c) { float v = xb[(size_t)c * N]; s += v * v; }
  sq[g] = s;
}

// fragment loader from global x (layout [C][N]) with point base col0
// VGPR0 (.x): K = k0 + 2*half ; VGPR1 (.y): K = k0 + 2*half + 1
__device__ inline v2f load_xfrag(const float* __restrict__ xb, int k0, int col0,
                                 int half, int l16) {
  v2f r;
  r.x = xb[(size_t)(k0 + 2 * half + 0) * N + col0 + l16];
  r.y = xb[(size_t)(k0 + 2 * half + 1) * N + col0 + l16];
  return r;
}

// fragment loader from LDS staging tile (layout [channel][32 cols])
__device__ inline v2f lds_frag(const float* st_, int k0, int colbase,
                               int half, int l16) {
  v2f r;
  r.x = st_[(k0 + 2 * half + 0) * 32 + colbase + l16];
  r.y = st_[(k0 + 2 * half + 1) * 32 + colbase + l16];
  return r;
}

__device__ inline v8f wmma4(v2f a, v2f b, v8f c) {
  return __builtin_amdgcn_wmma_f32_16x16x4_f32(false, a, false, b, (short)0, c,
                                               false, false);
}

// TDM: DMA a 2D tile (32 contiguous points x 64 channels, row stride N) into LDS.
// D# per CDNA5 ISA ch.8: group0 = {count/lds_addr/global_addr/type},
// group1 = {data_size, tensor dims, tile dims, dim0 stride}; groups 2/3 = 0 (2D).
__device__ inline void tdm_load_tile(unsigned int lds_byte_off, const void* gptr) {
  unsigned long long ga = (unsigned long long)(uintptr_t)gptr;
  u32x4 g0;
  g0.x = 1u;                                            // count=1 (valid), user mode
  g0.y = lds_byte_off;                                  // lds_addr (bytes)
  g0.z = (unsigned int)(ga & 0xFFFFFFFFu);              // global_addr[31:0]
  g0.w = (unsigned int)((ga >> 32) & 0x01FFFFFFu)       // global_addr[56:32]
         | (2u << 30);                                  // type=2 ("image")
  i32x8 g1;
  g1[0] = 0x00020000;                                   // data_size=2 (4 bytes)
  g1[1] = (int)((unsigned)(N & 0xFFFF) << 16);          // tensor_dim0 lo16
  g1[2] = (int)(((N >> 16) & 0xFFFF) | ((C & 0xFFFF) << 16));  // dim0 hi | dim1 lo
  g1[3] = (int)(((C >> 16) & 0xFFFF) | (32u << 16));    // dim1 hi | tile_dim0=32
  g1[4] = C;                                            // tile_dim1=64, tile_dim2=0
  g1[5] = N;                                            // tensor_dim0_stride lo32
  g1[6] = 0;
  g1[7] = 0;
  i32x4 z4 = {0, 0, 0, 0};
#if __clang_major__ >= 23
  i32x8 z8 = {0, 0, 0, 0, 0, 0, 0, 0};
  __builtin_amdgcn_tensor_load_to_lds(g0, g1, z4, z4, z8, 0);
#else
  __builtin_amdgcn_tensor_load_to_lds(g0, g1, z4, z4, 0);
#endif
}

// ---------------------------------------------------------------------------
// Kernel 2: tiled x·xᵀ via V_WMMA_F32_16X16X4_F32 + streaming per-row top-k.
// Block = 256 threads (8 waves); each wave owns a 16-row tile (128 rows/block).
// Per iteration: one TDM DMA stages the shared 32-col x 64-chan B-tile into
// LDS (all 8 waves use the identical tile -> 8x less VMEM traffic), then
// 4 independent WMMA accumulator chains per wave keep the matrix pipe full.
// Top-k: 256 lists (2 segments per row, all threads busy), merged at the end.
// Score: sq_j - 2*dot (sq_i constant per row, irrelevant to ordering).
// ---------------------------------------------------------------------------
__global__ void knn_kernel(const float* __restrict__ x, const float* __restrict__ sq,
                           int* __restrict__ idx) {
  __shared__ float stage[C * 32];      // 8 KB staging tile: [channel][col]
  __shared__ float ds2[128 * 32];
  __shared__ float tkd[256 * KNN];
  __shared__ int   tki[256 * KNN];

  const int b    = blockIdx.y;
  const int tid  = threadIdx.x;
  const int wave = tid >> 5;
  const int lane = tid & 31;
  const int half = lane >> 4;
  const int l16  = lane & 15;
  const int r0   = blockIdx.x * 128 + wave * 16;
  const float* xb = x + (size_t)b * C * N;
  const unsigned int stage_off = (unsigned int)(uintptr_t)&stage[0];

#pragma unroll
  for (int t = 0; t < KNN; ++t) { tkd[tid * KNN + t] = 3.4e38f; tki[tid * KNN + t] = 0; }
  float kth = 3.4e38f;

  // A fragments: this wave's 16 rows × all 64 channels (16 frags, reused N/32 times)
  v2f af[16];
#pragma unroll
  for (int kk = 0; kk < 16; ++kk) af[kk] = load_xfrag(xb, kk * 4, r0, half, l16);

  const int row = tid & 127;         // top-k row owned by this thread
  const int seg = tid >> 7;          // 0: cols 0-15 of tile, 1: cols 16-31
  float* td = &tkd[tid * KNN];
  int*   ti = &tki[tid * KNN];
  __syncthreads();

  for (int j0 = 0; j0 < N; j0 += 32) {
    if (wave == 0) {                 // one wave issues the DMA (wave-level op)
      if (j0 + 32 < N)               // warm GL2 for the following tile
        __builtin_prefetch(&xb[j0 + 32], 0, 1);
      tdm_load_tile(stage_off, (const void*)(xb + j0));
      __builtin_amdgcn_s_wait_tensorcnt(0);
    }
    __syncthreads();                 // tile visible to all waves

    // 4 independent accumulator chains, B fragments from LDS
    v8f c0e = {}, c0o = {}, c1e = {}, c1o = {};
#pragma unroll
    for (int kk = 0; kk < 16; kk += 2) {
      v2f b0e = lds_frag(stage, (kk + 0) * 4, 0,  half, l16);
      v2f b0o = lds_frag(stage, (kk + 1) * 4, 0,  half, l16);
      v2f b1e = lds_frag(stage, (kk + 0) * 4, 16, half, l16);
      v2f b1o = lds_frag(stage, (kk + 1) * 4, 16, half, l16);
      c0e = wmma4(af[kk + 0], b0e, c0e);
      c0o = wmma4(af[kk + 1], b0o, c0o);
      c1e = wmma4(af[kk + 0], b1e, c1e);
      c1o = wmma4(af[kk + 1], b1o, c1o);
    }
    float sq0 = sq[(size_t)b * N + j0 + l16];
    float sq1 = sq[(size_t)b * N + j0 + 16 + l16];
#pragma unroll
    for (int v = 0; v < 8; ++v) {
      int rloc = wave * 16 + v + half * 8;   // C layout: row = v (+8 for hi lanes)
      ds2[rloc * 32 + l16]      = sq0 - 2.0f * (c0e[v] + c0o[v]);
      ds2[rloc * 32 + 16 + l16] = sq1 - 2.0f * (c1e[v] + c1o[v]);
    }
    __syncthreads();

    {   // all 256 threads update their segment list
      const float* rowv = &ds2[row * 32 + seg * 16];
      const int colbase = j0 + seg * 16;
      for (int jj = 0; jj < 16; ++jj) {
        float vsc = rowv[jj];
        if (vsc < kth) {                     // register-cached kth: early reject
          int pos = KNN - 1;
          while (pos > 0 && td[pos - 1] > vsc) {  // ties keep earlier index
            td[pos] = td[pos - 1]; ti[pos] = ti[pos - 1]; --pos;
          }
          td[pos] = vsc; ti[pos] = colbase + jj;
          kth = td[KNN - 1];
        }
      }
    }
    __syncthreads();                 // lists done; stage may be overwritten
  }

  // merge the two segment lists per row (value, then index tie-break — matches top_k)
  if (tid < 128) {
    const float* tdA = &tkd[tid * KNN];
    const int*   tiA = &tki[tid * KNN];
    const float* tdB = &tkd[(tid + 128) * KNN];
    const int*   tiB = &tki[(tid + 128) * KNN];
    int ia = 0, ib = 0;
    int rowg = blockIdx.x * 128 + tid;
    int* dst = &idx[((size_t)b * N + rowg) * KNN];
#pragma unroll
    for (int t = 0; t < KNN; ++t) {
      float va = tdA[ia], vb = tdB[ib];
      bool pickA = (va < vb) || (va == vb && tiA[ia] < tiB[ib]);
      if (pickA) { dst[t] = tiA[ia]; ++ia; }
      else       { dst[t] = tiB[ib]; ++ib; }
    }
  }
}

// ---------------------------------------------------------------------------
// Kernel 3: q = xt·W1ᵀ, p = xt·(W2−W1)ᵀ  (both N×64×64 WMMA f32 GEMMs)
// h[b,n,k,o] = q[b,idx(n,k),o] + p[b,n,o]  replaces the 10.7 GFLOP edge GEMM.
// 4 independent accumulator chains (q/p × even/odd k-frags).
// ---------------------------------------------------------------------------
__global__ void pq_kernel(const float* __restrict__ x, const float* __restrict__ W,
                          float* __restrict__ p, float* __restrict__ q) {
  const int b    = blockIdx.y;
  const int tid  = threadIdx.x;
  const int wave = tid >> 5;
  const int lane = tid & 31;
  const int half = lane >> 4;
  const int l16  = lane & 15;
  const int r0   = blockIdx.x * 128 + wave * 16;
  const float* xb = x + (size_t)b * C * N;

  v2f af[16];
#pragma unroll
  for (int kk = 0; kk < 16; ++kk) af[kk] = load_xfrag(xb, kk * 4, r0, half, l16);

#pragma unroll
  for (int ot = 0; ot < 4; ++ot) {
    const int o = ot * 16 + l16;
    v8f aq0 = {}, aq1 = {}, ap0 = {}, ap1 = {};
#pragma unroll
    for (int kk = 0; kk < 16; kk += 2) {
      v2f bq0, bq1, bp0, bp1;
#pragma unroll
      for (int u = 0; u < 2; ++u) {
        int k0 = (kk + u) * 4 + 2 * half;
        float w10 = W[(size_t)o * 2 * C + k0 + 0];
        float w11 = W[(size_t)o * 2 * C + k0 + 1];
        v2f bq, bp;
        bq.x = w10;
        bq.y = w11;
        bp.x = W[(size_t)o * 2 * C + C + k0 + 0] - w10;
        bp.y = W[(size_t)o * 2 * C + C + k0 + 1] - w11;
        if (u == 0) { bq0 = bq; bp0 = bp; } else { bq1 = bq; bp1 = bp; }
      }
      aq0 = wmma4(af[kk + 0], bq0, aq0);
      ap0 = wmma4(af[kk + 0], bp0, ap0);
      aq1 = wmma4(af[kk + 1], bq1, aq1);
      ap1 = wmma4(af[kk + 1], bp1, ap1);
    }
#pragma unroll
    for (int v = 0; v < 8; ++v) {
      int n = r0 + v + half * 8;
      size_t base = ((size_t)b * N + n) * O + o;
      q[base] = aq0[v] + aq1[v];
      p[base] = ap0[v] + ap1[v];
    }
  }
}

// ---------------------------------------------------------------------------
// Kernel 4a: deterministic partial BN sums (per-channel sum, sumsq of h)
// Block = 4 rows × 64 channels; fixed strided row partition.
// ---------------------------------------------------------------------------
__global__ void stats_part_kernel(const float* __restrict__ p, const float* __restrict__ q,
                                  const int* __restrict__ idx, float* __restrict__ part) {
  __shared__ float s1s[4][64];
  __shared__ float s2s[4][64];
  const int tid = threadIdx.x;
  const int rg  = tid >> 6;
  const int o   = tid & 63;
  float s1 = 0.f, s2 = 0.f;
  for (int row = blockIdx.x * 4 + rg; row < B * N; row += NPART * 4) {
    int bb = row / N;
    float pv = p[(size_t)row * O + o];
#pragma unroll
    for (int k = 0; k < KNN; ++k) {
      int j = idx[(size_t)row * KNN + k];
      float h = q[((size_t)bb * N + j) * O + o] + pv;
      s1 += h; s2 += h * h;
    }
  }
  s1s[rg][o] = s1; s2s[rg][o] = s2;
  __syncthreads();
  if (rg == 0) {
    float a1 = s1s[0][o] + s1s[1][o] + s1s[2][o] + s1s[3][o];
    float a2 = s2s[0][o] + s2s[1][o] + s2s[2][o] + s2s[3][o];
    part[((size_t)blockIdx.x * O + o) * 2 + 0] = a1;
    part[((size_t)blockIdx.x * O + o) * 2 + 1] = a2;
  }
}

// Kernel 4b: fixed-order final reduction; fold BN into s,t:  hn = h*s + t
__global__ void stats_final_kernel(const float* __restrict__ part,
                                   const float* __restrict__ gamma,
                                   const float* __restrict__ beta,
                                   float* __restrict__ st) {
  int o = threadIdx.x;  // 64 threads
  float s1 = 0.f, s2 = 0.f;
  for (int i = 0; i < NPART; ++i) {
    s1 += part[((size_t)i * O + o) * 2 + 0];
    s2 += part[((size_t)i * O + o) * 2 + 1];
  }
  float cnt  = (float)B * (float)N * (float)KNN;
  float mean = s1 / cnt;
  float var  = s2 / cnt - mean * mean;
  float s    = rsqrtf(var + 1e-5f) * gamma[o];
  st[o]     = s;
  st[O + o] = beta[o] - mean * s;
}

// ---------------------------------------------------------------------------
// Kernel 5: fused gather + BN-affine + LeakyReLU + max over K + transpose
// Block = 4 points × 64 channels; q-row gathers are o-coalesced.
// ---------------------------------------------------------------------------
__global__ void final_kernel(const float* __restrict__ p, const float* __restrict__ q,
                             const int* __restrict__ idx, const float* __restrict__ st,
                             float* __restrict__ out) {
  const int tid = threadIdx.x;
  const int o   = tid & 63;
  const int nl  = tid >> 6;
  const int row = blockIdx.x * 4 + nl;          // global (b,n) row
  const int bb  = row / N, n = row % N;
  const float s = st[o], t = st[O + o];
  const float pv = p[(size_t)row * O + o];
  float m = -3.4e38f;
#pragma unroll
  for (int k = 0; k < KNN; ++k) {
    int j = idx[(size_t)row * KNN + k];
    float h  = q[((size_t)bb * N + j) * O + o] + pv;
    float hn = fmaf(h, s, t);
    float a  = hn >= 0.f ? hn : 0.2f * hn;
    m = fmaxf(m, a);
  }
  out[((size_t)bb * O + o) * N + n] = m;
}

// ---------------------------------------------------------------------------
extern "C" void kernel_launch(void* const* d_in, const int* in_sizes, int n_in,
                              void* d_out, int out_size, void* d_ws, size_t ws_size,
                              hipStream_t stream) {
  const float* x     = (const float*)d_in[0];
  const float* W     = (const float*)d_in[1];
  const float* gamma = (const float*)d_in[2];
  const float* beta  = (const float*)d_in[3];
  float* ws  = (float*)d_ws;
  float* sq  = ws + SQ_OFF;
  int*   idx = (int*)(ws + IDX_OFF);
  float* p   = ws + P_OFF;
  float* q   = ws + Q_OFF;
  float* prt = ws + PART_OFF;
  float* st  = ws + ST_OFF;
  float* out = (float*)d_out;

  sq_kernel<<<(B * N) / 256, 256, 0, stream>>>(x, sq);
  knn_kernel<<<dim3(N / 128, B), 256, 0, stream>>>(x, sq, idx);
  pq_kernel<<<dim3(N / 128, B), 256, 0, stream>>>(x, W, p, q);
  stats_part_kernel<<<NPART, 256, 0, stream>>>(p, q, idx, prt);
  stats_final_kernel<<<1, 64, 0, stream>>>(prt, gamma, beta, st);
  final_kernel<<<(B * N) / 4, 256, 0, stream>>>(p, q, idx, st, out);
}